// TtLlamaAttention_80436147519619
// MI455X (gfx1250) — compile-verified
//
#include <hip/hip_runtime.h>
#include <math.h>

// ---- problem constants (from reference) ----
#define B_    32
#define D_    4096
#define H_    32
#define HKV_  8
#define DH_   128
#define T_    4096
#define NQKV  6144   // H*DH + 2*HKV*DH  (q | k | v concatenated columns)
#define TROWS 16     // KV rows per async LDS tile

typedef __attribute__((ext_vector_type(2))) float v2f;
typedef __attribute__((ext_vector_type(8))) float v8f;

// ---------------------------------------------------------------------------
// Kernel 1: fused QKV projection, one wave per 16x16 output tile.
// C[m][n] = sum_k x[m][k] * W[k][n], W selected per column segment.
// Uses V_WMMA_F32_16X16X4_F32 (fp32 — workload is bandwidth bound, keep precision).
// ---------------------------------------------------------------------------
__global__ void qkv_gemm(const float* __restrict__ x,    // [32][4096]
                         const float* __restrict__ wq,   // [4096][4096]
                         const float* __restrict__ wk,   // [4096][1024]
                         const float* __restrict__ wv,   // [4096][1024]
                         float* __restrict__ qkv)        // [32][6144]
{
    const int ntile = blockIdx.x;        // 0..383
    const int mtile = blockIdx.y;        // 0..1
    const int lane  = threadIdx.x;       // 0..31 (wave32)
    const int nloc  = lane & 15;
    const int hk    = lane >> 4;         // half-K selector per A/B fragment layout

    const float* w; int ldw; int col0;
    if (ntile < 256)      { w = wq; ldw = 4096; col0 = ntile * 16; }
    else if (ntile < 320) { w = wk; ldw = 1024; col0 = (ntile - 256) * 16; }
    else                  { w = wv; ldw = 1024; col0 = (ntile - 320) * 16; }

    const int m = mtile * 16 + nloc;          // A row this lane supplies
    const int n = col0 + nloc;                // B col this lane supplies
    const float* arow = x + (size_t)m * D_;

    v8f c = {0.f, 0.f, 0.f, 0.f, 0.f, 0.f, 0.f, 0.f};
    #pragma unroll 8
    for (int k0 = 0; k0 < D_; k0 += 4) {
        const int ka = k0 + hk * 2;
        v2f a; a.x = arow[ka];                 a.y = arow[ka + 1];
        v2f b; b.x = w[(size_t)ka * ldw + n];  b.y = w[(size_t)(ka + 1) * ldw + n];
        c = __builtin_amdgcn_wmma_f32_16x16x4_f32(false, a, false, b,
                                                  (short)0, c, false, false);
    }

    const int outcol = ntile * 16 + nloc;
    #pragma unroll
    for (int r = 0; r < 8; ++r) {
        const int mr = mtile * 16 + r + hk * 8;   // D layout: M = r (+8 for hi lanes)
        qkv[(size_t)mr * NQKV + outcol] = c[r];
    }
}

// ---------------------------------------------------------------------------
// Kernel 2: RoPE for q and k. Rotation coefficients read from rot_mat input
// (block-diagonal 2x2: c = R[2i][2i], s = R[2i][2i+1]).
// ---------------------------------------------------------------------------
__global__ void rope_kernel(const float* __restrict__ qkv,   // [32][6144]
                            const float* __restrict__ rot,   // [1][32][128][128]
                            float* __restrict__ qrot,        // [32][32][128]
                            float* __restrict__ krot)        // [32][8][128]
{
    const int NQ = B_ * H_ * (DH_ / 2);     // 65536
    const int NK = B_ * HKV_ * (DH_ / 2);   // 16384
    int tid = blockIdx.x * blockDim.x + threadIdx.x;
    if (tid >= NQ + NK) return;

    const bool isq = tid < NQ;
    const int idx  = isq ? tid : tid - NQ;
    const int pair = idx & 63;              // DH/2 = 64 pairs
    const int head = idx >> 6;
    const int b    = isq ? head / H_ : head / HKV_;
    const int hh   = isq ? head % H_ : head % HKV_;

    const float* r = rot + (size_t)b * DH_ * DH_;
    const int e = pair * 2;
    const float cc = r[(size_t)e * DH_ + e];
    const float ss = r[(size_t)e * DH_ + e + 1];

    const size_t src = (size_t)b * NQKV + (isq ? 0 : (H_ * DH_)) + (size_t)hh * DH_;
    const float x0 = qkv[src + e];
    const float x1 = qkv[src + e + 1];

    float* dst = isq ? (qrot + ((size_t)b * H_ + hh) * DH_)
                     : (krot + ((size_t)b * HKV_ + hh) * DH_);
    dst[e]     = cc * x0 - ss * x1;
    dst[e + 1] = ss * x0 + cc * x1;
}

// ---------------------------------------------------------------------------
// CDNA5 async HBM -> LDS helpers (ASYNCcnt-tracked; no VGPR round-trip).
// Per-lane LDS byte address = low 32 bits of flat shared pointer (ISA 10.2:
// LDS aperture address truncation).
// ---------------------------------------------------------------------------
__device__ __forceinline__ unsigned int lds_off(const void* p) {
    return (unsigned int)(uintptr_t)p;
}

// Copy one KV tile (TROWS contiguous rows of 128 floats each, 8KB per matrix,
// linear in memory) into LDS. 256 threads * 2 chunks * 16B = 8KB per matrix.
// Issues exactly 4 async instructions per wave (2 per matrix).
__device__ __forceinline__ void async_copy_tile(const float* gK, const float* gV,
                                                unsigned int lK, unsigned int lV,
                                                int tid)
{
    #pragma unroll
    for (int c = 0; c < 2; ++c) {
        const unsigned int byteOff = (unsigned int)(tid + c * 256) * 16u; // 0..8176
        unsigned long long ga = (unsigned long long)(uintptr_t)gK + byteOff;
        unsigned int       la = lK + byteOff;
        asm volatile("global_load_async_to_lds_b128 %0, %1, off"
                     :: "v"(la), "v"(ga) : "memory");
        ga = (unsigned long long)(uintptr_t)gV + byteOff;
        la = lV + byteOff;
        asm volatile("global_load_async_to_lds_b128 %0, %1, off"
                     :: "v"(la), "v"(ga) : "memory");
    }
}

__device__ __forceinline__ void wait_async_le4() {
    asm volatile("s_wait_asynccnt 0x4" ::: "memory");
}
__device__ __forceinline__ void wait_async_0() {
    asm volatile("s_wait_asynccnt 0x0" ::: "memory");
}

// ---------------------------------------------------------------------------
// Kernel 3: GQA flash attention, one workgroup (8 waves) per (b, kv-head).
// Masked region (t > start_pos) contributes exactly 0 after softmax, so we
// iterate t in [0, start_pos], substituting new k/v at t == start_pos
// (the reference's functional cache update; inputs never mutated).
// KV rows [0, start_pos & ~15) stream HBM->LDS via double-buffered
// global_load_async_to_lds_b128; the tail rows (including t==start_pos)
// use direct loads. Per row: wave32 butterfly dot, online softmax,
// LDS combine across the 8 waves.
// ---------------------------------------------------------------------------
__global__ void attn_kernel(const float* __restrict__ qrot,   // [32][32][128]
                            const float* __restrict__ krot,   // [32][8][128]
                            const float* __restrict__ qkv,    // v at col 5120
                            const float* __restrict__ kcache, // [32][8][4096][128]
                            const float* __restrict__ vcache, // [32][8][4096][128]
                            const int*   __restrict__ start_pos,
                            float* __restrict__ attn_out)     // [32][4096]
{
    const int kv   = blockIdx.x;        // 0..7
    const int b    = blockIdx.y;        // 0..31
    const int tid  = threadIdx.x;       // 0..255
    const int lane = tid & 31;
    const int wave = tid >> 5;          // 0..7

    __shared__ float bufK[2][TROWS][DH_];   // 16KB
    __shared__ float bufV[2][TROWS][DH_];   // 16KB
    __shared__ float red_m[8][4];
    __shared__ float red_l[8][4];
    __shared__ float red_o[8][4][DH_];      // 16KB

    const int sp = start_pos[0];
    const int nT = sp + 1;

    float4 qreg[4];
    #pragma unroll
    for (int g = 0; g < 4; ++g) {
        const float* qp = qrot + ((size_t)b * H_ + kv * 4 + g) * DH_ + lane * 4;
        qreg[g] = *(const float4*)qp;
    }

    const float* kbase = kcache + (size_t)(b * HKV_ + kv) * T_ * DH_;
    const float* vbase = vcache + (size_t)(b * HKV_ + kv) * T_ * DH_;
    const float* knew  = krot + ((size_t)b * HKV_ + kv) * DH_;
    const float* vnew  = qkv + (size_t)b * NQKV + (H_ * DH_ + HKV_ * DH_) + kv * DH_;

    float  m[4], l[4];
    float4 acc[4];
    #pragma unroll
    for (int g = 0; g < 4; ++g) {
        m[g] = -INFINITY; l[g] = 0.f;
        acc[g] = make_float4(0.f, 0.f, 0.f, 0.f);
    }
    const float scale = 0.08838834764831845f;   // 1/sqrt(128)

    // --- online softmax update for one (k-row, v-row) pair, all 4 q heads ---
    auto update = [&](float4 k4, float4 v4) {
        #pragma unroll
        for (int g = 0; g < 4; ++g) {
            float p = k4.x * qreg[g].x + k4.y * qreg[g].y
                    + k4.z * qreg[g].z + k4.w * qreg[g].w;
            #pragma unroll
            for (int off = 16; off > 0; off >>= 1)
                p += __shfl_xor(p, off, 32);        // wave32 butterfly: full dot
            const float s    = p * scale;
            const float mnew = fmaxf(m[g], s);
            const float corr = __expf(m[g] - mnew); // exp(-inf)=0 on first row
            const float pe   = __expf(s - mnew);
            l[g] = l[g] * corr + pe;
            acc[g].x = acc[g].x * corr + pe * v4.x;
            acc[g].y = acc[g].y * corr + pe * v4.y;
            acc[g].z = acc[g].z * corr + pe * v4.z;
            acc[g].w = acc[g].w * corr + pe * v4.w;
            m[g] = mnew;
        }
    };

    // --- tiled region [0, nFull): async double-buffered HBM->LDS stream ---
    const int nFull    = sp & ~(TROWS - 1);
    const int numTiles = nFull / TROWS;

    const unsigned int lK0 = lds_off(&bufK[0][0][0]);
    const unsigned int lK1 = lds_off(&bufK[1][0][0]);
    const unsigned int lV0 = lds_off(&bufV[0][0][0]);
    const unsigned int lV1 = lds_off(&bufV[1][0][0]);

    if (numTiles > 0)
        async_copy_tile(kbase, vbase, lK0, lV0, tid);

    for (int i = 0; i < numTiles; ++i) {
        const bool more = (i + 1) < numTiles;
        if (more) {
            const size_t off = (size_t)(i + 1) * TROWS * DH_;
            async_copy_tile(kbase + off, vbase + off,
                            ((i + 1) & 1) ? lK1 : lK0,
                            ((i + 1) & 1) ? lV1 : lV0, tid);
        }
        if (more) wait_async_le4(); else wait_async_0();  // this wave's tile done
        __syncthreads();                                   // all waves' tiles done

        const int buf = i & 1;
        #pragma unroll
        for (int r = 0; r < 2; ++r) {                      // 8 waves x 2 rows = 16
            const int row = wave * 2 + r;
            const float4 k4 = *(const float4*)&bufK[buf][row][lane * 4];
            const float4 v4 = *(const float4*)&bufV[buf][row][lane * 4];
            update(k4, v4);
        }
        __syncthreads();            // done reading before buffer is overwritten
    }

    // --- tail rows [nFull, sp], direct loads; t==sp takes new k/v ---
    for (int t = nFull + wave; t < nT; t += 8) {
        const float* kr = (t == sp) ? knew : (kbase + (size_t)t * DH_);
        const float* vr = (t == sp) ? vnew : (vbase + (size_t)t * DH_);
        const float4 k4 = *(const float4*)(kr + lane * 4);
        const float4 v4 = *(const float4*)(vr + lane * 4);
        update(k4, v4);
    }

    // --- combine the 8 per-wave flash states through LDS ---
    #pragma unroll
    for (int g = 0; g < 4; ++g) {
        if (lane == 0) { red_m[wave][g] = m[g]; red_l[wave][g] = l[g]; }
        float* o = &red_o[wave][g][lane * 4];
        o[0] = acc[g].x; o[1] = acc[g].y; o[2] = acc[g].z; o[3] = acc[g].w;
    }
    __syncthreads();

    for (int idx = tid; idx < 4 * DH_; idx += 256) {
        const int g = idx >> 7, d = idx & (DH_ - 1);
        float M = -INFINITY;
        #pragma unroll
        for (int w = 0; w < 8; ++w) M = fmaxf(M, red_m[w][g]);
        float L = 0.f, O = 0.f;
        #pragma unroll
        for (int w = 0; w < 8; ++w) {
            const float f = __expf(red_m[w][g] - M);
            L += red_l[w][g] * f;
            O += red_o[w][g][d] * f;
        }
        attn_out[(size_t)b * (H_ * DH_) + (kv * 4 + g) * DH_ + d] = O / L;
    }
}

// ---------------------------------------------------------------------------
// Kernel 4: output projection (32x4096) @ (4096x4096) with fp32 WMMA.
// ---------------------------------------------------------------------------
__global__ void out_gemm(const float* __restrict__ a,    // [32][4096] attn out
                         const float* __restrict__ wo,   // [4096][4096]
                         float* __restrict__ out)        // [32][4096]
{
    const int ntile = blockIdx.x;        // 0..255
    const int mtile = blockIdx.y;        // 0..1
    const int lane  = threadIdx.x;
    const int nloc  = lane & 15;
    const int hk    = lane >> 4;

    const int m = mtile * 16 + nloc;
    const int n = ntile * 16 + nloc;
    const float* arow = a + (size_t)m * (H_ * DH_);

    v8f c = {0.f, 0.f, 0.f, 0.f, 0.f, 0.f, 0.f, 0.f};
    #pragma unroll 8
    for (int k0 = 0; k0 < H_ * DH_; k0 += 4) {
        const int ka = k0 + hk * 2;
        v2f av; av.x = arow[ka];                  av.y = arow[ka + 1];
        v2f bv; bv.x = wo[(size_t)ka * D_ + n];   bv.y = wo[(size_t)(ka + 1) * D_ + n];
        c = __builtin_amdgcn_wmma_f32_16x16x4_f32(false, av, false, bv,
                                                  (short)0, c, false, false);
    }

    #pragma unroll
    for (int r = 0; r < 8; ++r) {
        const int mr = mtile * 16 + r + hk * 8;
        out[(size_t)mr * D_ + ntile * 16 + nloc] = c[r];
    }
}

// ---------------------------------------------------------------------------
extern "C" void kernel_launch(void* const* d_in, const int* in_sizes, int n_in,
                              void* d_out, int out_size, void* d_ws, size_t ws_size,
                              hipStream_t stream) {
    const float* x   = (const float*)d_in[0];   // (1,1,32,4096)
    const float* wq  = (const float*)d_in[1];   // (4096,4096)
    const float* wk  = (const float*)d_in[2];   // (4096,1024)
    const float* wv  = (const float*)d_in[3];   // (4096,1024)
    const float* wo  = (const float*)d_in[4];   // (4096,4096)
    const float* rot = (const float*)d_in[5];   // (1,32,128,128)
    // d_in[6] attn_mask: not needed — masked probs are exactly 0 in fp32
    const float* kc  = (const float*)d_in[7];   // (32,8,4096,128)
    const float* vc  = (const float*)d_in[8];   // (32,8,4096,128)
    const int*   sp  = (const int*)d_in[9];     // scalar start_pos

    // workspace layout (floats): qkv | qrot | krot | attn_out  (~1.9 MB)
    float* ws    = (float*)d_ws;
    float* qkv   = ws;                        // 32*6144 = 196608
    float* qrot  = ws + 196608;               // 32*32*128 = 131072
    float* krot  = ws + 327680;               // 32*8*128 = 32768
    float* aout  = ws + 360448;               // 32*4096 = 131072

    qkv_gemm <<<dim3(NQKV / 16, 2), 32, 0, stream>>>(x, wq, wk, wv, qkv);
    rope_kernel<<<(B_*H_*64 + B_*HKV_*64 + 255) / 256, 256, 0, stream>>>(qkv, rot, qrot, krot);
    attn_kernel<<<dim3(HKV_, B_), 256, 0, stream>>>(qrot, krot, qkv, kc, vc, sp, aout);
    out_gemm <<<dim3(D_ / 16, 2), 32, 0, stream>>>(aout, wo, (float*)d_out);
}